// Intra_66185446031427
// MI455X (gfx1250) — compile-verified
//
#include <hip/hip_runtime.h>
#include <math.h>
#include <stdint.h>

typedef __attribute__((ext_vector_type(2))) float v2f;
typedef __attribute__((ext_vector_type(4))) float v4f;
typedef __attribute__((ext_vector_type(8))) float v8f;
typedef unsigned int u32x4 __attribute__((ext_vector_type(4)));
typedef int i32x4 __attribute__((ext_vector_type(4)));
typedef int i32x8 __attribute__((ext_vector_type(8)));

#define BATCH 8
#define LSEQ  2048
#define HDIM  512
#define MT    32            // query rows per workgroup
#define NKV   64            // kv rows per iteration
#define QS    (HDIM + 4)    // padded row stride (floats): TDM pads +2 DW per 256 DW
#define SSTR  (NKV + 4)     // padded row stride for S/P tile (68)
#define SCALE 0.04419417382415922f   // 1/sqrt(512)
#define NTHREADS 256

#if defined(__has_builtin)
#if __has_builtin(__builtin_amdgcn_tensor_load_to_lds)
#define HAVE_TDM 1
#endif
#endif

// padded column offset (TDM inserts 2 DWORDs after every 256 DWORDs)
__device__ __forceinline__ int pcol(int c) { return c + ((c >> 8) << 1); }

#ifdef HAVE_TDM
// Issue one TDM DMA: rows x 512 f32 tile (row-major, stride 512) -> LDS,
// padded +2 DWORDs per 256 DWORDs (effective LDS row stride = 516 floats).
__device__ __forceinline__ void tdm_load_tile(const float* gptr, unsigned lds_off,
                                              int rows) {
    const unsigned long long ga = (unsigned long long)(uintptr_t)gptr;
    u32x4 g0;
    g0[0] = 1u;                                   // count=1, user descriptor
    g0[1] = lds_off;                              // lds_addr (bytes)
    g0[2] = (unsigned)(ga & 0xffffffffull);       // global_addr[31:0]
    g0[3] = (unsigned)((ga >> 32) & 0x1ffffffull) // global_addr[56:32]
          | (2u << 30);                           // type = 2 (image)
    i32x8 g1;
    g1[0] = (int)((2u << 16)      // data_size = 4B
                | (1u << 20)      // pad_enable
                | (7u << 22)      // pad_interval: every 256 DWORDs
                | (1u << 25));    // pad_amount: 2 DWORDs
    g1[1] = (int)(512u << 16);    // tensor_dim0[15:0]=512
    g1[2] = (int)((unsigned)rows << 16);  // tensor_dim1[15:0]=rows
    g1[3] = (int)(512u << 16);    // tile_dim0 = 512
    g1[4] = rows;                 // tile_dim1 = rows, tile_dim2 = 0 (2D)
    g1[5] = 512;                  // tensor_dim0_stride[31:0] = 512 elements
    g1[6] = 0;
    g1[7] = 0;
    const i32x4 gz4 = {0, 0, 0, 0};
    const i32x8 gz8 = {0, 0, 0, 0, 0, 0, 0, 0};
    __builtin_amdgcn_tensor_load_to_lds(g0, g1, gz4, gz4, gz8, 0);
}
#endif

__launch_bounds__(NTHREADS, 1)
__global__ void fused_attn_residual_f32(const float* __restrict__ A,
                                        const float* __restrict__ Bm,
                                        float* __restrict__ Out) {
    __shared__ float Qs[MT * QS];        // 66048 B
    __shared__ float Ks[NKV * QS];       // 132096 B
    __shared__ float Ss[MT * SSTR];      // S tile -> P tile (8704 B)
    __shared__ float pmax[8 * MT];
    __shared__ float psum[8 * MT];
    __shared__ float row_max[MT];
    __shared__ float row_sum[MT];
    __shared__ float row_scl[MT];

    const int tid  = threadIdx.x;
    const int lane = tid & 31;
    const int w    = tid >> 5;                 // wave id 0..7
    const int batch = blockIdx.x >> 6;         // 64 row-blocks per batch
    const int q0    = (blockIdx.x & 63) * MT;

    const float* Aq = A  + ((size_t)batch * LSEQ + q0) * HDIM;
    const float* Bb = Bm + (size_t)batch * LSEQ * HDIM;

    // ---- stage Q tile (32 x 512) into LDS via Tensor Data Mover ----
#ifdef HAVE_TDM
    if (w == 0) {
        tdm_load_tile(Aq, (unsigned)(uintptr_t)&Qs[0], MT);
        __builtin_amdgcn_s_wait_tensorcnt(0);
    }
#else
    for (int i = tid; i < MT * (HDIM / 4); i += NTHREADS) {
        const int r = i >> 7, c4 = i & 127;
        *(v4f*)&Qs[r * QS + pcol(c4 * 4)] = *(const v4f*)(Aq + r * HDIM + c4 * 4);
    }
#endif
    if (tid < MT) {
        row_max[tid] = -__builtin_inff();
        row_sum[tid] = 0.0f;
        row_scl[tid] = 0.0f;
    }

    // O accumulators: this wave's 64-wide H slice, 2 M-tiles x 4 H-tiles of 16x16
    v8f acc[2][4];
    #pragma unroll
    for (int i = 0; i < 2; ++i)
        #pragma unroll
        for (int j = 0; j < 4; ++j)
            acc[i][j] = (v8f){0.f,0.f,0.f,0.f,0.f,0.f,0.f,0.f};

    // S-tile assignment: S is 32x64 = 2x4 tiles of 16x16, one per wave
    const int mi  = w >> 2;                    // 0..1
    const int ni  = w & 3;                     // 0..3
    const int l15 = lane & 15;
    const int kh  = (lane >> 4) * 2;           // K-pair offset within 4-wide K step
    const int arow = mi * 16 + l15;            // Q row for A fragment
    const int brow = ni * 16 + l15;            // K row (= S column) for B fragment
    const int srow = tid & 31;                 // softmax: row
    const int sseg = tid >> 5;                 // softmax: 8-col segment

    for (int kv0 = 0; kv0 < LSEQ; kv0 += NKV) {
        __syncthreads();   // previous iteration done reading Ks / Ss

        // ---- stage K/V tile (64 x 512) via TDM ----
        const float* Bt = Bb + (size_t)kv0 * HDIM;
#ifdef HAVE_TDM
        if (w == 0) {
            tdm_load_tile(Bt, (unsigned)(uintptr_t)&Ks[0], NKV);
            __builtin_amdgcn_s_wait_tensorcnt(0);
        }
#else
        for (int i = tid; i < NKV * (HDIM / 4); i += NTHREADS) {
            const int r = i >> 7, c4 = i & 127;
            *(v4f*)&Ks[r * QS + pcol(c4 * 4)] = *(const v4f*)(Bt + r * HDIM + c4 * 4);
        }
#endif
        __syncthreads();

        // ---- S = Q * K^T : one 16x16 tile per wave, K-dim = 512 ----
        v8f s = (v8f){0.f,0.f,0.f,0.f,0.f,0.f,0.f,0.f};
        #pragma unroll
        for (int half = 0; half < 2; ++half) {
            const float* qb = &Qs[arow * QS + half * 258 + kh];
            const float* kb = &Ks[brow * QS + half * 258 + kh];
            #pragma unroll 4
            for (int k0 = 0; k0 < 256; k0 += 4) {
                const v2f af = *(const v2f*)(qb + k0);
                const v2f bf = *(const v2f*)(kb + k0);
                s = __builtin_amdgcn_wmma_f32_16x16x4_f32(false, af, false, bf,
                                                          (short)0, s, false, false);
            }
        }
        // write scaled S tile (D layout: lane half selects M+8)
        {
            const int col   = ni * 16 + l15;
            const int rbase = mi * 16 + ((lane >> 4) << 3);
            #pragma unroll
            for (int r = 0; r < 8; ++r)
                Ss[(rbase + r) * SSTR + col] = s[r] * SCALE;
        }
        __syncthreads();

        // ---- online softmax, 8 threads per row ----
        {   // phase 1: partial max over this thread's 8 columns
            float pm = -__builtin_inff();
            #pragma unroll
            for (int j = 0; j < 8; ++j)
                pm = fmaxf(pm, Ss[srow * SSTR + sseg * 8 + j]);
            pmax[sseg * MT + srow] = pm;
        }
        __syncthreads();
        if (tid < MT) {   // phase 2: new row max + rescale factor
            const float mo = row_max[tid];
            float mn = mo;
            #pragma unroll
            for (int sg = 0; sg < 8; ++sg) mn = fmaxf(mn, pmax[sg * MT + tid]);
            row_max[tid] = mn;
            row_scl[tid] = __expf(mo - mn);     // 0 on first iteration
        }
        __syncthreads();
        {   // phase 3: exponentiate in place, partial sums
            const float mn = row_max[srow];
            float ps = 0.0f;
            #pragma unroll
            for (int j = 0; j < 8; ++j) {
                const int idx = srow * SSTR + sseg * 8 + j;
                const float p = __expf(Ss[idx] - mn);
                Ss[idx] = p;
                ps += p;
            }
            psum[sseg * MT + srow] = ps;
        }
        __syncthreads();
        if (tid < MT) {   // phase 4: running denominator
            float t = 0.0f;
            #pragma unroll
            for (int sg = 0; sg < 8; ++sg) t += psum[sg * MT + tid];
            row_sum[tid] = row_sum[tid] * row_scl[tid] + t;
        }
        __syncthreads();

        // ---- O-GEMM: rescale then O += P * V (V = K tile), 64-wide H slice ----
        // Stage all P (A) fragments explicitly: pa[mi2][k] covers K-dim 64.
        v2f pa[2][16];
        #pragma unroll
        for (int mi2 = 0; mi2 < 2; ++mi2) {
            const int prow = mi2 * 16 + l15;
            #pragma unroll
            for (int k = 0; k < 16; ++k)
                pa[mi2][k] = *(const v2f*)&Ss[prow * SSTR + k * 4 + kh];
        }
        // rescale accumulators by exp(m_old - m_new)
        #pragma unroll
        for (int mi2 = 0; mi2 < 2; ++mi2) {
            const int rb = mi2 * 16 + ((lane >> 4) << 3);
            float scl[8];
            #pragma unroll
            for (int r = 0; r < 8; ++r) scl[r] = row_scl[rb + r];
            #pragma unroll
            for (int hi = 0; hi < 4; ++hi)
                #pragma unroll
                for (int r = 0; r < 8; ++r) acc[mi2][hi][r] *= scl[r];
        }
        // hi outer, k inner; both mi2 consume the same B fragment immediately
        #pragma unroll
        for (int hi = 0; hi < 4; ++hi) {
            const int hoff = pcol(w * 64 + hi * 16 + l15);
            #pragma unroll
            for (int k = 0; k < 16; ++k) {
                v2f bf;
                bf.x = Ks[(k * 4 + kh) * QS + hoff];
                bf.y = Ks[(k * 4 + kh + 1) * QS + hoff];
                acc[0][hi] = __builtin_amdgcn_wmma_f32_16x16x4_f32(
                    false, pa[0][k], false, bf, (short)0, acc[0][hi], false, false);
                acc[1][hi] = __builtin_amdgcn_wmma_f32_16x16x4_f32(
                    false, pa[1][k], false, bf, (short)0, acc[1][hi], false, false);
            }
        }
    }

    __syncthreads();

    // ---- epilogue: out = O / row_sum + Q (residual) ----
    #pragma unroll
    for (int mi2 = 0; mi2 < 2; ++mi2) {
        const int rb = mi2 * 16 + ((lane >> 4) << 3);
        float inv[8];
        #pragma unroll
        for (int r = 0; r < 8; ++r) inv[r] = 1.0f / row_sum[rb + r];
        #pragma unroll
        for (int hi = 0; hi < 4; ++hi) {
            const int hcol = w * 64 + hi * 16 + l15;
            const int hoff = pcol(hcol);
            #pragma unroll
            for (int r = 0; r < 8; ++r) {
                const int row = rb + r;
                const float v = acc[mi2][hi][r] * inv[r] + Qs[row * QS + hoff];
                Out[((size_t)batch * LSEQ + q0 + row) * HDIM + hcol] = v;
            }
        }
    }
}

extern "C" void kernel_launch(void* const* d_in, const int* in_sizes, int n_in,
                              void* d_out, int out_size, void* d_ws, size_t ws_size,
                              hipStream_t stream) {
    (void)in_sizes; (void)n_in; (void)d_ws; (void)ws_size; (void)out_size;
    const float* a = (const float*)d_in[0];
    const float* b = (const float*)d_in[1];
    float* out = (float*)d_out;

    const int blocks = BATCH * (LSEQ / MT);   // 8 * 64 = 512
    fused_attn_residual_f32<<<dim3(blocks), dim3(NTHREADS), 0, stream>>>(a, b, out);
}